// ProposalLayer_87840671137915
// MI455X (gfx1250) — compile-verified
//
#include <hip/hip_runtime.h>
#include <hip/hip_bf16.h>
#include <math.h>

#define NPROP 8192
#define NWORDS 256          // NPROP / 32
#define NPOST 2000
#define MIN_SZ 16.0f
#define IMG 800.0f
#define THRESH 0.7f

#define AS3 __attribute__((address_space(3)))
typedef int v4i __attribute__((vector_size(16)));   // matches builtin param type

// ---------------------------------------------------------------------------
// 0) zero the valid counter
// ---------------------------------------------------------------------------
__global__ void init_kernel(unsigned* vcount) { *vcount = 0u; }

// ---------------------------------------------------------------------------
// 1) decode + clip + validity + sort-key construction
//    key = (~sortable(score_or_-inf) << 32) | index  -> ascending u64 sort
//    gives score-descending with index-ascending tiebreak (matches argsort).
// ---------------------------------------------------------------------------
__global__ void decode_kernel(const float4* __restrict__ locs,
                              const float*  __restrict__ scores,
                              float4* __restrict__ boxes4,
                              unsigned long long* __restrict__ keys,
                              unsigned* __restrict__ vcount) {
  int i = blockIdx.x * blockDim.x + threadIdx.x;
  if (i >= NPROP) return;
  float4 l = locs[i];                 // cy, cx, h, w
  float y1 = l.x - 0.5f * l.z;
  float x1 = l.y - 0.5f * l.w;
  float y2 = l.x + 0.5f * l.z;
  float x2 = l.y + 0.5f * l.w;
  bool valid = ((y2 - y1) > MIN_SZ) && ((x2 - x1) > MIN_SZ);
  float4 b;
  b.x = fminf(fmaxf(y1, 0.0f), IMG);
  b.y = fminf(fmaxf(x1, 0.0f), IMG);
  b.z = fminf(fmaxf(y2, 0.0f), IMG);
  b.w = fminf(fmaxf(x2, 0.0f), IMG);
  boxes4[i] = b;

  float kf = valid ? scores[i] : -__builtin_huge_valf();
  unsigned u = __float_as_uint(kf);
  unsigned sortable = (u & 0x80000000u) ? ~u : (u | 0x80000000u); // ascending-float order
  unsigned desc = ~sortable;                                      // descending-float order
  keys[i] = (((unsigned long long)desc) << 32) | (unsigned)i;
  if (valid) atomicAdd(vcount, 1u);
}

// ---------------------------------------------------------------------------
// 2) one-block bitonic sort of 8192 u64 keys in 64KB dynamic LDS (wave32 WGP)
// ---------------------------------------------------------------------------
__global__ void sort_kernel(const unsigned long long* __restrict__ keys,
                            unsigned* __restrict__ order) {
  extern __shared__ unsigned long long ks[];
  const int tid = threadIdx.x;
  for (int i = tid; i < NPROP; i += 1024) ks[i] = keys[i];
  __syncthreads();
  for (int k = 2; k <= NPROP; k <<= 1) {
    for (int j = k >> 1; j > 0; j >>= 1) {
      for (int t = tid; t < NPROP; t += 1024) {
        int p = t ^ j;
        if (p > t) {
          unsigned long long a = ks[t];
          unsigned long long b = ks[p];
          bool asc = ((t & k) == 0);
          if ((a > b) == asc) { ks[t] = b; ks[p] = a; }
        }
      }
      __syncthreads();
    }
  }
  for (int i = tid; i < NPROP; i += 1024) order[i] = (unsigned)(ks[i] & 0xFFFFFFFFu);
}

// ---------------------------------------------------------------------------
// 3) gather boxes into sorted order (coalesced producer for the IoU kernel)
// ---------------------------------------------------------------------------
__global__ void gather_kernel(const float4* __restrict__ boxes4,
                              const unsigned* __restrict__ order,
                              float4* __restrict__ sboxes) {
  int i = blockIdx.x * blockDim.x + threadIdx.x;
  if (i < NPROP) sboxes[i] = boxes4[order[i]];
}

// ---------------------------------------------------------------------------
// 4) pairwise IoU suppression bitmask. Block = (32 cols, 16 rows).
//    Col/row boxes are staged into LDS via GLOBAL_LOAD_ASYNC_TO_LDS_B128
//    (ASYNCcnt path); wave0 waits s_wait_asynccnt before the block barrier.
//    Blocks entirely below the diagonal (chunkEnd < row0) are skipped; the
//    scan only consults words w >= i>>5, so those words need not be written.
// ---------------------------------------------------------------------------
__global__ void iou_kernel(const float4* __restrict__ sb,
                           unsigned* __restrict__ mask) {
  __shared__ float4 lcol[32];
  __shared__ float4 lrow[16];

  const int chunk = blockIdx.x;
  const int chunkEnd = chunk * 32 + 31;
  const int row0 = blockIdx.y * 16;
  if (chunkEnd < row0) return;   // uniform: whole block strictly below diagonal

  const int lane = threadIdx.x;
  if (threadIdx.y == 0) {
    // stage 32 column boxes (512B) + 16 row boxes (256B) into LDS, async
    __builtin_amdgcn_global_load_async_to_lds_b128(
        (v4i*)(sb + (chunk * 32 + lane)),
        (AS3 v4i*)&lcol[lane], 0, 0);
    if (lane < 16) {
      __builtin_amdgcn_global_load_async_to_lds_b128(
          (v4i*)(sb + (row0 + lane)),
          (AS3 v4i*)&lrow[lane], 0, 0);
    }
    asm volatile("s_wait_asynccnt 0x0" ::: "memory");
  }
  __syncthreads();

  const int row = row0 + threadIdx.y;
  if (chunkEnd < row) return;    // wave-level skip: word never consulted

  float4 a = lrow[threadIdx.y];  // uniform per wave
  float4 c = lcol[lane];
  float areaA = (a.z - a.x) * (a.w - a.y);
  float areaC = (c.z - c.x) * (c.w - c.y);
  float iy = fmaxf(0.0f, fminf(a.z, c.z) - fmaxf(a.x, c.x));
  float ix = fmaxf(0.0f, fminf(a.w, c.w) - fmaxf(a.y, c.y));
  float inter = iy * ix;
  float uni = areaA + areaC - inter;
  float iou = (uni > 0.0f) ? (inter / uni) : 0.0f;
  int col = chunk * 32 + lane;
  bool pred = (col > row) && (iou > THRESH);
  unsigned long long bal = __ballot(pred);   // wave32: low 32 bits
  if (lane == 0)
    mask[(size_t)row * NWORDS + chunk] = (unsigned)bal;
}

// ---------------------------------------------------------------------------
// 5) greedy serial scan: single wave32, remv bitmap in LDS (ds_or atomics),
//    global_prefetch_b8 on upcoming mask rows to hide fetch latency.
//    Only words w >= i>>5 are valid/consulted (upper triangle).
// ---------------------------------------------------------------------------
__global__ void scan_kernel(const unsigned* __restrict__ mask,
                            const float4* __restrict__ sboxes,
                            const unsigned* __restrict__ vcount,
                            float4* __restrict__ out) {
  __shared__ unsigned remv[NWORDS];
  const int lane = threadIdx.x;
  const int V = (int)(*vcount);   // validity sorts to a prefix

  // init remv: bit set (suppressed) for sorted index i >= V
#pragma unroll
  for (int s = 0; s < 8; ++s) {
    int w = lane + 32 * s;
    int base = w * 32;
    unsigned m;
    if (V <= base)            m = 0xFFFFFFFFu;
    else if (V >= base + 32)  m = 0u;
    else                      m = 0xFFFFFFFFu << (V - base);
    remv[w] = m;
  }
  __syncthreads();

  int nkept = 0;
  for (int i = 0; i < NPROP; ++i) {
    if (i >= V || nkept >= NPOST) break;
    // prefetch an upcoming mask row (gfx1250 global_prefetch_b8)
    if (i + 24 < V)
      __builtin_prefetch(mask + (size_t)(i + 24) * NWORDS + lane * 8, 0, 3);

    const int iw = i >> 5;
    unsigned wv = remv[iw];              // uniform broadcast LDS load
    if ((wv >> (i & 31)) & 1u) continue; // suppressed

    if (lane == 0) out[nkept] = sboxes[i];
    nkept++;                              // uniform across the wave

    const unsigned* row = mask + (size_t)i * NWORDS;
#pragma unroll
    for (int s = 0; s < 8; ++s) {
      int w = lane + 32 * s;              // coalesced 128B per step
      if (w >= iw) {                      // only upper-triangle words exist
        unsigned m = row[w];
        if (m) atomicOr(&remv[w], m);     // ds_or_b32
      }
    }
    __builtin_amdgcn_wave_barrier();
  }

  // zero-fill remaining output slots (d_out is poisoned by the harness)
  for (int o = nkept + lane; o < NPOST; o += 32)
    out[o] = make_float4(0.0f, 0.0f, 0.0f, 0.0f);
}

// ---------------------------------------------------------------------------
extern "C" void kernel_launch(void* const* d_in, const int* in_sizes, int n_in,
                              void* d_out, int out_size, void* d_ws, size_t ws_size,
                              hipStream_t stream) {
  const float4* locs   = (const float4*)d_in[0];  // (8192,4) cy,cx,h,w
  const float*  scores = (const float*)d_in[1];   // (8192,)
  float4*       out    = (float4*)d_out;          // (2000,4)

  char* ws = (char*)d_ws;
  unsigned*           mask   = (unsigned*)(ws);                       // 8,388,608 B
  float4*             boxes4 = (float4*)(ws + 8388608);               //   131,072 B
  float4*             sboxes = (float4*)(ws + 8388608 + 131072);      //   131,072 B
  unsigned long long* keys   = (unsigned long long*)(ws + 8650752);   //    65,536 B
  unsigned*           order  = (unsigned*)(ws + 8716288);             //    32,768 B
  unsigned*           vcount = (unsigned*)(ws + 8749056);             //         4 B

  init_kernel<<<1, 1, 0, stream>>>(vcount);
  decode_kernel<<<NPROP / 256, 256, 0, stream>>>(locs, scores, boxes4, keys, vcount);
  sort_kernel<<<1, 1024, NPROP * sizeof(unsigned long long), stream>>>(keys, order);
  gather_kernel<<<NPROP / 256, 256, 0, stream>>>(boxes4, order, sboxes);

  dim3 g(NWORDS, NPROP / 16);
  dim3 b(32, 16);
  iou_kernel<<<g, b, 0, stream>>>(sboxes, mask);

  scan_kernel<<<1, 32, 0, stream>>>(mask, sboxes, vcount, out);
}